// Method_GCN_Modified_39419209843176
// MI455X (gfx1250) — compile-verified
//
#include <hip/hip_runtime.h>
#include <math.h>

typedef __attribute__((ext_vector_type(2))) float v2f;
typedef __attribute__((ext_vector_type(8))) float v8f;

#define BN_EPS 1e-5f

// ---------------------------------------------------------------------------
// row_ptr[i] = lower_bound(edge_row, i)  (edge_row is sorted)
// ---------------------------------------------------------------------------
__global__ void build_rowptr_kernel(const int* __restrict__ erow, int E,
                                    int* __restrict__ rowptr, int n) {
  int i = blockIdx.x * blockDim.x + threadIdx.x;
  if (i > n) return;
  int lo = 0, hi = E;
  while (lo < hi) {
    int mid = (lo + hi) >> 1;
    if (erow[mid] < i) lo = mid + 1; else hi = mid;
  }
  rowptr[i] = lo;
}

// ---------------------------------------------------------------------------
// WT[c*K + k] = (c < C) ? W[k*C + c] : 0   — weight transpose with zero pad.
// Makes both GEMM fragments K-contiguous (single b64 load, no predication).
// ---------------------------------------------------------------------------
__global__ void transpose_pad_kernel(const float* __restrict__ W,
                                     float* __restrict__ WT,
                                     int K, int C, int Cpad) {
  const int i = blockIdx.x * blockDim.x + threadIdx.x;
  if (i >= Cpad * K) return;
  const int c = i / K;
  const int k = i - c * K;
  WT[i] = (c < C) ? W[k * C + c] : 0.0f;
}

// ---------------------------------------------------------------------------
// out[nrows x OSTRIDE] = A[nrows x K] * W  (W given transposed: WT[OSTRIDE x K])
// One wave32 computes TWO stacked 16x16 tiles (32 rows) sharing each B frag.
// blockDim.x = 32 * (OSTRIDE/16). Inner loop: 3x global_load_b64 + 2x WMMA,
// no branches, EXEC all-ones (WMMA requirement).
// Fragment layouts per ISA: A lane l -> row l&15, K = (l>>4)*2..+1;
// B mirrors with column on the lane; C/D lane l -> col l&15, rows (l>>4)*8+j.
// ---------------------------------------------------------------------------
template<int K, int OSTRIDE>
__global__ void gemm_wmma_kernel(const float* __restrict__ A,
                                 const float* __restrict__ WT,
                                 float* __restrict__ out, int nrows) {
  const int lane = threadIdx.x & 31;
  const int wave = threadIdx.x >> 5;
  const int row0 = blockIdx.x * 32;
  if (row0 + 32 > nrows) return;              // N % 32 == 0 for this problem
  const int m  = lane & 15;
  const int kh = (lane >> 4) << 1;            // lanes 0-15: K=0,1  lanes 16-31: K=2,3
  const int n  = wave * 16 + m;
  const float* arow0 = A + (row0 + m) * K + kh;
  const float* arow1 = arow0 + 16 * K;
  const float* brow  = WT + n * K + kh;

  v8f c0 = {}, c1 = {};
  #pragma unroll 4
  for (int k0 = 0; k0 < K; k0 += 4) {
    const v2f b  = *(const v2f*)(brow  + k0);
    const v2f a0 = *(const v2f*)(arow0 + k0);
    const v2f a1 = *(const v2f*)(arow1 + k0);
    c0 = __builtin_amdgcn_wmma_f32_16x16x4_f32(false, a0, false, b,
                                               (short)0, c0, false, false);
    c1 = __builtin_amdgcn_wmma_f32_16x16x4_f32(false, a1, false, b,
                                               (short)0, c1, false, false);
  }

  const int mb = (lane >> 4) << 3;            // C/D: lanes 16-31 hold rows M+8
  float* orow0 = out + (row0 + mb) * OSTRIDE + n;
  float* orow1 = orow0 + 16 * OSTRIDE;
  #pragma unroll
  for (int j = 0; j < 8; ++j) {
    orow0[j * OSTRIDE] = c0[j];
    orow1[j * OSTRIDE] = c1[j];
  }
}

// ---------------------------------------------------------------------------
// CSR SpMM: out[i][f] = sum_{e in [ptr[i],ptr[i+1])} val[e] * sup[col[e]][f]
// One wave32 per node. Edges fetched cooperatively (coalesced, 32 at a time)
// then broadcast via shuffles; feature gathers are b64 (float2 per lane).
// Deterministic (sequential edge order, no atomics); gathers L2-resident.
// ---------------------------------------------------------------------------
template<int STRIDE, int F>   // F even; sup row stride STRIDE; out packed F
__global__ void spmm_kernel(const float* __restrict__ sup,
                            const int* __restrict__ rowptr,
                            const int* __restrict__ cols,
                            const float* __restrict__ vals,
                            float* __restrict__ out, int n) {
  const int node = blockIdx.x * (blockDim.x >> 5) + (threadIdx.x >> 5);
  if (node >= n) return;
  const int lane = threadIdx.x & 31;
  const int f0 = lane * 2;
  const bool act = (f0 < F);
  const int e0 = rowptr[node];
  const int e1 = rowptr[node + 1];

  float ax = 0.0f, ay = 0.0f;
  for (int eb = e0; eb < e1; eb += 32) {
    const int ee = eb + lane;
    const int   cL = (ee < e1) ? cols[ee] : 0;
    const float vL = (ee < e1) ? vals[ee] : 0.0f;
    const int cnt = min(32, e1 - eb);
    for (int j = 0; j < cnt; ++j) {
      const int   c = __shfl(cL, j, 32);
      const float v = __shfl(vL, j, 32);
      if (act) {
        const v2f s = *(const v2f*)(sup + c * STRIDE + f0);
        ax += v * s.x;
        ay += v * s.y;
      }
    }
  }
  if (act) {
    v2f r; r.x = ax; r.y = ay;
    *(v2f*)(out + node * F + f0) = r;
  }
}

// ---------------------------------------------------------------------------
// Deterministic per-feature batch stats: stats[2f]=sum, stats[2f+1]=sumsq
// ---------------------------------------------------------------------------
__global__ void bn_stats_kernel(const float* __restrict__ h, int n, int M,
                                float* __restrict__ stats) {
  const int f = blockIdx.x;
  float s = 0.0f, s2 = 0.0f;
  for (int i = threadIdx.x; i < n; i += blockDim.x) {
    const float x = h[i * M + f];
    s += x; s2 += x * x;
  }
  __shared__ float sh[256], sh2[256];
  sh[threadIdx.x] = s; sh2[threadIdx.x] = s2;
  __syncthreads();
  for (int o = blockDim.x >> 1; o > 0; o >>= 1) {
    if (threadIdx.x < (unsigned)o) {
      sh[threadIdx.x]  += sh[threadIdx.x + o];
      sh2[threadIdx.x] += sh2[threadIdx.x + o];
    }
    __syncthreads();
  }
  if (threadIdx.x == 0) { stats[2 * f] = sh[0]; stats[2 * f + 1] = sh2[0]; }
}

// BatchNorm (training-mode batch stats, biased var) + ELU, in place.
__global__ void bn_elu_kernel(float* __restrict__ h, const float* __restrict__ stats,
                              const float* __restrict__ gamma, const float* __restrict__ beta,
                              int n, int M) {
  const int i = blockIdx.x * blockDim.x + threadIdx.x;
  if (i >= n * M) return;
  const int f = i % M;
  const float inv_n = 1.0f / (float)n;
  const float mean  = stats[2 * f] * inv_n;
  const float var   = stats[2 * f + 1] * inv_n - mean * mean;
  const float x = (h[i] - mean) * rsqrtf(var + BN_EPS) * gamma[f] + beta[f];
  h[i] = x > 0.0f ? x : expm1f(x);
}

// ---------------------------------------------------------------------------
// Row-wise log_softmax over 40 logits; one wave32 per node, shuffle reductions.
// ---------------------------------------------------------------------------
__global__ void log_softmax_kernel(const float* __restrict__ in,
                                   float* __restrict__ out, int n) {
  const int node = blockIdx.x * (blockDim.x >> 5) + (threadIdx.x >> 5);
  if (node >= n) return;
  const int lane = threadIdx.x & 31;
  const float* r = in + node * 40;
  const bool  has2 = (lane + 32) < 40;
  const float x0 = r[lane];
  const float x1 = has2 ? r[lane + 32] : -__builtin_inff();
  float m = fmaxf(x0, x1);
  #pragma unroll
  for (int o = 16; o > 0; o >>= 1) m = fmaxf(m, __shfl_xor(m, o, 32));
  float s = expf(x0 - m) + (has2 ? expf(x1 - m) : 0.0f);
  #pragma unroll
  for (int o = 16; o > 0; o >>= 1) s += __shfl_xor(s, o, 32);
  const float lse = m + logf(s);
  float* o0 = out + node * 40;
  o0[lane] = x0 - lse;
  if (has2) o0[lane + 32] = x1 - lse;
}

// ---------------------------------------------------------------------------
extern "C" void kernel_launch(void* const* d_in, const int* in_sizes, int n_in,
                              void* d_out, int out_size, void* d_ws, size_t ws_size,
                              hipStream_t stream) {
  const float* x     = (const float*)d_in[0];
  const int*   erow  = (const int*)d_in[1];
  const int*   ecol  = (const int*)d_in[2];
  const float* eval_ = (const float*)d_in[3];
  const float* W1 = (const float*)d_in[4];
  const float* W2 = (const float*)d_in[5];
  const float* W3 = (const float*)d_in[6];
  const float* W4 = (const float*)d_in[7];
  const float* g1 = (const float*)d_in[8];
  const float* b1 = (const float*)d_in[9];
  const float* g2 = (const float*)d_in[10];
  const float* b2 = (const float*)d_in[11];
  const float* g3 = (const float*)d_in[12];
  const float* b3 = (const float*)d_in[13];

  const int N = in_sizes[0] / 512;   // 100000
  const int E = in_sizes[1];         // 3200000

  // Workspace: rowptr | stats | WT1..WT4 | bufS | bufH   (~52 MB)
  char* ws = (char*)d_ws;
  size_t off = 0;
  auto alloc = [&](size_t bytes) -> void* {
    void* p = ws + off;
    off = (off + bytes + 255) & ~(size_t)255;
    return p;
  };
  int*   rowptr = (int*)  alloc((size_t)(N + 1) * sizeof(int));
  float* stats  = (float*)alloc(2 * 64 * sizeof(float));
  float* WT1 = (float*)alloc((size_t)64 * 512 * sizeof(float));
  float* WT2 = (float*)alloc((size_t)32 * 64  * sizeof(float));
  float* WT3 = (float*)alloc((size_t)16 * 32  * sizeof(float));
  float* WT4 = (float*)alloc((size_t)48 * 16  * sizeof(float));
  float* bufS = (float*)alloc((size_t)N * 64 * sizeof(float));  // support (padded L4)
  float* bufH = (float*)alloc((size_t)N * 64 * sizeof(float));  // hidden / logits

  // One-time prep (tiny): CSR row pointers + transposed/padded weights.
  build_rowptr_kernel<<<(N + 256) / 256, 256, 0, stream>>>(erow, E, rowptr, N);
  transpose_pad_kernel<<<(64 * 512 + 255) / 256, 256, 0, stream>>>(W1, WT1, 512, 64, 64);
  transpose_pad_kernel<<<(32 * 64  + 255) / 256, 256, 0, stream>>>(W2, WT2, 64,  32, 32);
  transpose_pad_kernel<<<(16 * 32  + 255) / 256, 256, 0, stream>>>(W3, WT3, 32,  16, 16);
  transpose_pad_kernel<<<(48 * 16  + 255) / 256, 256, 0, stream>>>(W4, WT4, 16,  40, 48);

  const int gtiles = (N + 31) / 32;   // N % 32 == 0 here (100000/32 = 3125)
  const int sblks  = (N + 7) / 8;     // 8 waves (nodes) per 256-thread block

  // ---- Layer 1: 512 -> 64 ----
  gemm_wmma_kernel<512, 64><<<gtiles, 4 * 32, 0, stream>>>(x, WT1, bufS, N);
  spmm_kernel<64, 64><<<sblks, 256, 0, stream>>>(bufS, rowptr, ecol, eval_, bufH, N);
  bn_stats_kernel<<<64, 256, 0, stream>>>(bufH, N, 64, stats);
  bn_elu_kernel<<<(N * 64 + 255) / 256, 256, 0, stream>>>(bufH, stats, g1, b1, N, 64);

  // ---- Layer 2: 64 -> 32 ----
  gemm_wmma_kernel<64, 32><<<gtiles, 2 * 32, 0, stream>>>(bufH, WT2, bufS, N);
  spmm_kernel<32, 32><<<sblks, 256, 0, stream>>>(bufS, rowptr, ecol, eval_, bufH, N);
  bn_stats_kernel<<<32, 256, 0, stream>>>(bufH, N, 32, stats);
  bn_elu_kernel<<<(N * 32 + 255) / 256, 256, 0, stream>>>(bufH, stats, g2, b2, N, 32);

  // ---- Layer 3: 32 -> 16 ----
  gemm_wmma_kernel<32, 16><<<gtiles, 1 * 32, 0, stream>>>(bufH, WT3, bufS, N);
  spmm_kernel<16, 16><<<sblks, 256, 0, stream>>>(bufS, rowptr, ecol, eval_, bufH, N);
  bn_stats_kernel<<<16, 256, 0, stream>>>(bufH, N, 16, stats);
  bn_elu_kernel<<<(N * 16 + 255) / 256, 256, 0, stream>>>(bufH, stats, g3, b3, N, 16);

  // ---- Layer 4: 16 -> 40 (WT4 zero-padded to 48 rows; whole 16-wide tiles) ----
  gemm_wmma_kernel<16, 48><<<gtiles, 3 * 32, 0, stream>>>(bufH, WT4, bufS, N);
  spmm_kernel<48, 40><<<sblks, 256, 0, stream>>>(bufS, rowptr, ecol, eval_, bufH, N);
  log_softmax_kernel<<<sblks, 256, 0, stream>>>(bufH, (float*)d_out, N);
}